// EquivLayerNorm_88751204205256
// MI455X (gfx1250) — compile-verified
//
#include <hip/hip_runtime.h>

// EquivLayerNorm for MI455X (gfx1250, wave32).
// Pass 1: WMMA-based per-node reductions (sum s, sum s^2, sum v^2) + f32 global atomics per segment.
// Pass 2: finalize per-segment params (mean, 1/var, 1/vmean).
// Pass 3: streaming apply with float4 loads/stores.

typedef float v2f __attribute__((ext_vector_type(2)));
typedef float v8f __attribute__((ext_vector_type(8)));

#define SDIM   128
#define VDIM   64
#define VROW   192      // 3 * VDIM floats per node in v
#define EPS_F  1e-6f
#define B_CAP  2048     // capacity for segment accumulators in workspace

// ---------------------------------------------------------------------------
// Pass 1: one wave32 handles a tile of 16 nodes.
// B-matrix trick: D = Ones(16x4) x B(4x16) accumulates column sums; with A all
// ones the result is independent of the exact K-slot mapping, so we just need
// lanes L and L+16 (same column n = L&15) to load the node's 4 consecutive
// features as two float2's.
// ---------------------------------------------------------------------------
__global__ __launch_bounds__(256)
void eln_stats_kernel(const float* __restrict__ s,
                      const float* __restrict__ v,
                      const int*   __restrict__ batch,
                      float*       __restrict__ acc,   // [B_CAP][4]: S1, S2, V1, cnt
                      int N, int numTiles)
{
    int gtid = blockIdx.x * blockDim.x + threadIdx.x;
    int wave = gtid >> 5;          // one tile of 16 nodes per wave
    int lane = gtid & 31;
    if (wave >= numTiles) return;  // wave-uniform branch: EXEC stays all-ones

    int base  = wave * 16;
    int nloc  = lane & 15;                 // column (node within tile)
    int koff  = (lane >> 4) * 2;           // which feature pair this lane owns
    int node  = base + nloc;
    float msk = (node < N) ? 1.0f : 0.0f;  // zero-pad tail tile without branching
    int ncl   = (node < N) ? node : (N - 1);

    const float* srow = s + (size_t)ncl * SDIM + koff;
    const float* vrow = v + (size_t)ncl * VROW + koff;

    v2f a; a[0] = 1.0f; a[1] = 1.0f;       // all-ones A matrix (16x4)
    v8f accS = {};   // column sums of s
    v8f accQ = {};   // column sums of s^2
    v8f accV = {};   // column sums of v^2

    // scalar branch: 128 features = 32 chunks of K=4
    for (int k0 = 0; k0 < SDIM; k0 += 4) {
        v2f b = *(const v2f*)(srow + k0);
        b[0] *= msk; b[1] *= msk;
        v2f bq; bq[0] = b[0] * b[0]; bq[1] = b[1] * b[1];
        accS = __builtin_amdgcn_wmma_f32_16x16x4_f32(false, a, false, b,
                                                     (short)0, accS, false, false);
        accQ = __builtin_amdgcn_wmma_f32_16x16x4_f32(false, a, false, bq,
                                                     (short)0, accQ, false, false);
    }
    // vector branch: 192 values = 48 chunks of K=4 (sum of squares only)
    for (int k0 = 0; k0 < VROW; k0 += 4) {
        v2f b = *(const v2f*)(vrow + k0);
        v2f bq; bq[0] = b[0] * b[0] * msk; bq[1] = b[1] * b[1] * msk;
        accV = __builtin_amdgcn_wmma_f32_16x16x4_f32(false, a, false, bq,
                                                     (short)0, accV, false, false);
    }

    // D[0][n] (element 0, lanes 0..15) holds the full reduction for node n.
    if (lane < 16 && node < N) {
        int seg = batch[node];
        float* p = acc + (size_t)seg * 4;
        atomicAdd(p + 0, accS[0]);
        atomicAdd(p + 1, accQ[0]);
        atomicAdd(p + 2, accV[0]);
        atomicAdd(p + 3, 1.0f);
    }
}

// ---------------------------------------------------------------------------
// Pass 2: per-segment parameters.  var = segmean(mean_f(s^2)) - m^2 (exact
// rewrite of the reference's two-stage formula since segmean(mean_f(s)) == m).
// ---------------------------------------------------------------------------
__global__ __launch_bounds__(256)
void eln_finalize_kernel(const float* __restrict__ acc,
                         float*       __restrict__ params,
                         const int*   __restrict__ nseg_p)
{
    int i = blockIdx.x * blockDim.x + threadIdx.x;
    int nseg = *nseg_p;
    if (i >= nseg) return;
    float s1 = acc[i * 4 + 0];
    float s2 = acc[i * 4 + 1];
    float vv = acc[i * 4 + 2];
    float cn = fmaxf(acc[i * 4 + 3], 1.0f);
    float m   = s1 / (cn * (float)SDIM);
    float msq = s2 / (cn * (float)SDIM);
    float var = fmaxf(msq - m * m, EPS_F);
    float vmn = fmaxf(vv / (cn * (float)VDIM), EPS_F);
    params[i * 4 + 0] = m;
    params[i * 4 + 1] = 1.0f / var;
    params[i * 4 + 2] = 1.0f / vmn;
    params[i * 4 + 3] = 0.0f;
}

// ---------------------------------------------------------------------------
// Pass 3a: sout = (s - m) * (1/var) * w + b, float4-vectorized streaming.
// ---------------------------------------------------------------------------
__global__ __launch_bounds__(256)
void eln_apply_s_kernel(const float* __restrict__ s,
                        const float* __restrict__ w,
                        const float* __restrict__ bias,
                        const int*   __restrict__ batch,
                        const float* __restrict__ params,
                        float*       __restrict__ out,
                        int N)
{
    int idx = blockIdx.x * blockDim.x + threadIdx.x;   // one float4 chunk
    int chunks = N * (SDIM / 4);
    if (idx >= chunks) return;
    int n  = idx >> 5;              // SDIM/4 == 32 chunks per row
    int f4 = (idx & 31) * 4;
    int seg = batch[n];
    float m  = params[seg * 4 + 0];
    float iv = params[seg * 4 + 1];
    float4 sv = ((const float4*)s)[idx];
    float4 wv = *(const float4*)(w + f4);
    float4 bv = *(const float4*)(bias + f4);
    float4 o;
    o.x = (sv.x - m) * iv * wv.x + bv.x;
    o.y = (sv.y - m) * iv * wv.y + bv.y;
    o.z = (sv.z - m) * iv * wv.z + bv.z;
    o.w = (sv.w - m) * iv * wv.w + bv.w;
    ((float4*)out)[idx] = o;
}

// ---------------------------------------------------------------------------
// Pass 3b: vout = v * (1/vmean), float4-vectorized streaming.
// ---------------------------------------------------------------------------
__global__ __launch_bounds__(256)
void eln_apply_v_kernel(const float* __restrict__ v,
                        const int*   __restrict__ batch,
                        const float* __restrict__ params,
                        float*       __restrict__ out,
                        int N)
{
    int idx = blockIdx.x * blockDim.x + threadIdx.x;   // one float4 chunk
    int chunks = N * (VROW / 4);                       // 48 chunks per node
    if (idx >= chunks) return;
    int n = idx / 48;
    int seg = batch[n];
    float ivm = params[seg * 4 + 2];
    float4 vv = ((const float4*)v)[idx];
    float4 o;
    o.x = vv.x * ivm; o.y = vv.y * ivm; o.z = vv.z * ivm; o.w = vv.w * ivm;
    ((float4*)out)[idx] = o;
}

// ---------------------------------------------------------------------------
extern "C" void kernel_launch(void* const* d_in, const int* in_sizes, int n_in,
                              void* d_out, int out_size, void* d_ws, size_t ws_size,
                              hipStream_t stream)
{
    const float* s      = (const float*)d_in[0];
    const float* v      = (const float*)d_in[1];
    const float* w      = (const float*)d_in[2];
    const float* bias   = (const float*)d_in[3];
    const int*   batch  = (const int*)d_in[4];
    const int*   nseg_p = (const int*)d_in[5];

    int N = in_sizes[0] / SDIM;

    float* acc    = (float*)d_ws;          // B_CAP * 4 floats
    float* params = acc + B_CAP * 4;       // B_CAP * 4 floats

    // Accumulators must be zero every call (ws is poisoned once, not per-call).
    hipMemsetAsync(acc, 0, (size_t)B_CAP * 4 * sizeof(float), stream);

    dim3 blk(256);

    int numTiles = (N + 15) / 16;
    int statThreads = numTiles * 32;
    eln_stats_kernel<<<(statThreads + 255) / 256, blk, 0, stream>>>(
        s, v, batch, acc, N, numTiles);

    eln_finalize_kernel<<<(B_CAP + 255) / 256, blk, 0, stream>>>(
        acc, params, nseg_p);

    int sChunks = N * (SDIM / 4);
    eln_apply_s_kernel<<<(sChunks + 255) / 256, blk, 0, stream>>>(
        s, w, bias, batch, params, (float*)d_out, N);

    int vChunks = N * (VROW / 4);
    float* outv = (float*)d_out + (size_t)N * SDIM;
    eln_apply_v_kernel<<<(vChunks + 255) / 256, blk, 0, stream>>>(
        v, batch, params, outv, N);
}